// RandomWalkPE_84851373899971
// MI455X (gfx1250) — compile-verified
//
#include <hip/hip_runtime.h>

typedef __attribute__((ext_vector_type(2))) float v2f;
typedef __attribute__((ext_vector_type(8))) float v8f;

#define NN 4096
#define NEDGE 65536
#define BM 128          // block tile M
#define BNB 256         // block tile N
#define BK 32           // block tile K
#define LDA (BK + 4)    // As row stride: 36 dwords (144B, 16B aligned, bank-clean)
#define LDBP 544        // Bp row stride in dwords: 512 + 32 (shifts upper lane-half banks)

// ---------------- zero (float4) ----------------
__global__ __launch_bounds__(256) void zero_f4(float4* __restrict__ p, int n4) {
    int i = blockIdx.x * 256 + threadIdx.x;
    if (i < n4) p[i] = make_float4(0.f, 0.f, 0.f, 0.f);
}

// ---------------- adjacency scatter ----------------
__global__ __launch_bounds__(256) void scatter_edges(const int* __restrict__ ei,
                                                     float* __restrict__ A) {
    int e = blockIdx.x * 256 + threadIdx.x;
    if (e < NEDGE) {
        int s = ei[e];
        int t = ei[NEDGE + e];
        atomicAdd(&A[(size_t)s * NN + t], 1.0f);
        atomicAdd(&A[(size_t)t * NN + s], 1.0f);
    }
}

// ---------------- row normalize: T[i,:] = A[i,:] / max(rowsum,1) ----------------
__global__ __launch_bounds__(256) void row_normalize(float* __restrict__ A) {
    int row = blockIdx.x;
    int tid = threadIdx.x;
    float* r = A + (size_t)row * NN;
    float s = 0.f;
    for (int j = tid; j < NN; j += 256) s += r[j];
    for (int off = 16; off > 0; off >>= 1) s += __shfl_down(s, off, 32);
    __shared__ float red[8];
    if ((tid & 31) == 0) red[tid >> 5] = s;
    __syncthreads();
    __shared__ float invdeg;
    if (tid == 0) {
        float tot = 0.f;
        for (int w = 0; w < 8; ++w) tot += red[w];
        invdeg = 1.0f / fmaxf(tot, 1.0f);
    }
    __syncthreads();
    float inv = invdeg;
    for (int j = tid; j < NN; j += 256) r[j] *= inv;
}

// ---------------- C = A * B, 4096^3 f32, WMMA 16x16x4 ----------------
// Block tile 128x256, BK=32, 8 waves in 2(M)x4(N), wave tile 64x64 (4x4 WMMA accums).
// A fragment: lanes 0-15 hold K={kk,kk+1} of rows m0..m0+15; lanes 16-31 hold K={kk+2,kk+3}.
// B in LDS is K-pair interleaved: (k,n) at Bp[k>>1][n*2 + (k&1)] so each frag is one b64.
__global__ __launch_bounds__(256) void gemm_wmma_f32(const float* __restrict__ A,
                                                     const float* __restrict__ B,
                                                     float* __restrict__ C) {
    __shared__ float As[BM][LDA];              // 18 KB
    __shared__ float Bp[(BK / 2) * LDBP];      // ~34.8 KB
    const int tid  = threadIdx.x;
    const int lane = tid & 31;
    const int wave = tid >> 5;        // 0..7
    const int wm = wave >> 2;         // 0..1  (64-row strip)
    const int wn = wave & 3;          // 0..3  (64-col strip)
    const int l15 = lane & 15;
    const int halfSel = lane >> 4;    // 0: K={0,1}, 1: K={2,3} (A/B); rows v / v+8 (C)
    const int blockM = blockIdx.y * BM;
    const int blockN = blockIdx.x * BNB;

    v8f acc[4][4];
    const v8f vzero = {0.f, 0.f, 0.f, 0.f, 0.f, 0.f, 0.f, 0.f};
#pragma unroll
    for (int mi = 0; mi < 4; ++mi)
#pragma unroll
        for (int ni = 0; ni < 4; ++ni) acc[mi][ni] = vzero;

    // A staging: 128x32 tile, thread loads 4 float4 (rows t>>3 + 32r, col (t&7)*4)
    const int aCol = (tid & 7) * 4;
    const int aRow = tid >> 3;
    // B staging: 32x256 tile, thread handles row pairs k=(t>>6)*2+8r at cols (t&63)*4..+3
    const int bCol = (tid & 63) * 4;
    const int bK0  = (tid >> 6) * 2;

    for (int kt = 0; kt < NN; kt += BK) {
#pragma unroll
        for (int r = 0; r < 4; ++r) {
            const float4 v = *(const float4*)(&A[(size_t)(blockM + aRow + 32 * r) * NN + kt + aCol]);
            *(float4*)&As[aRow + 32 * r][aCol] = v;
        }
#pragma unroll
        for (int r = 0; r < 4; ++r) {
            const int k = bK0 + 8 * r;   // even row of the pair
            const float4 va = *(const float4*)(&B[(size_t)(kt + k) * NN + blockN + bCol]);
            const float4 vb = *(const float4*)(&B[(size_t)(kt + k + 1) * NN + blockN + bCol]);
            float* dst = &Bp[(k >> 1) * LDBP + bCol * 2];
            const float4 lo = make_float4(va.x, vb.x, va.y, vb.y);
            const float4 hi = make_float4(va.z, vb.z, va.w, vb.w);
            *(float4*)(dst + 0) = lo;
            *(float4*)(dst + 4) = hi;
        }
        __syncthreads();

#pragma unroll
        for (int kk = 0; kk < BK; kk += 4) {
            const int kbase = kk + halfSel * 2;       // even
            v2f afrag[4];
            v2f bfrag[4];
#pragma unroll
            for (int mi = 0; mi < 4; ++mi)
                afrag[mi] = *(const v2f*)&As[wm * 64 + mi * 16 + l15][kbase];
#pragma unroll
            for (int ni = 0; ni < 4; ++ni)
                bfrag[ni] = *(const v2f*)&Bp[(kbase >> 1) * LDBP + (wn * 64 + ni * 16 + l15) * 2];
#pragma unroll
            for (int mi = 0; mi < 4; ++mi)
#pragma unroll
                for (int ni = 0; ni < 4; ++ni)
                    acc[mi][ni] = __builtin_amdgcn_wmma_f32_16x16x4_f32(
                        false, afrag[mi], false, bfrag[ni],
                        (short)0, acc[mi][ni], false, false);
        }
        __syncthreads();
    }

#pragma unroll
    for (int mi = 0; mi < 4; ++mi) {
#pragma unroll
        for (int ni = 0; ni < 4; ++ni) {
            const int baseRow = blockM + wm * 64 + mi * 16 + halfSel * 8;
            const int col = blockN + wn * 64 + ni * 16 + l15;
#pragma unroll
            for (int v = 0; v < 8; ++v)
                C[(size_t)(baseRow + v) * NN + col] = acc[mi][ni][v];
        }
    }
}

// ---------------- diag extract: d[i] = M[i,i] ----------------
__global__ __launch_bounds__(256) void diag_extract(const float* __restrict__ M,
                                                    float* __restrict__ d) {
    int i = blockIdx.x * 256 + threadIdx.x;
    if (i < NN) d[i] = M[(size_t)i * NN + i];
}

// ---------------- d[i] = sum_j X[i,j] * Y[j,i]  (deterministic, tiled) ----------------
__global__ __launch_bounds__(256) void diag_dot(const float* __restrict__ X,
                                                const float* __restrict__ Y,
                                                float* __restrict__ dout) {
    __shared__ float Yt[64][65];
    const int tid  = threadIdx.x;
    const int lane = tid & 31;
    const int wave = tid >> 5;       // 0..7 -> 8 rows each
    const int i0 = blockIdx.x * 64;

    float partial[8];
#pragma unroll
    for (int q = 0; q < 8; ++q) partial[q] = 0.f;

    for (int tj = 0; tj < NN; tj += 64) {
        __syncthreads();
        // load Y[tj+jj][i0+ii] -> Yt[ii][jj]  (coalesced on ii)
#pragma unroll
        for (int s = 0; s < 16; ++s) {
            const int idx = tid + s * 256;
            const int jj = idx >> 6;
            const int ii = idx & 63;
            Yt[ii][jj] = Y[(size_t)(tj + jj) * NN + i0 + ii];
        }
        __syncthreads();
#pragma unroll
        for (int q = 0; q < 8; ++q) {
            const int r = wave * 8 + q;
            const float* xr = X + (size_t)(i0 + r) * NN + tj;
            partial[q] += xr[lane] * Yt[r][lane] + xr[lane + 32] * Yt[r][lane + 32];
        }
    }
#pragma unroll
    for (int q = 0; q < 8; ++q) {
        float s = partial[q];
        for (int off = 16; off > 0; off >>= 1) s += __shfl_down(s, off, 32);
        if (lane == 0) dout[i0 + wave * 8 + q] = s;
    }
}

// ---------------- out[i,d] = b[d] + sum_k diag[k][i] * W[k][d] ----------------
__global__ __launch_bounds__(256) void pe_project(const float* __restrict__ diag,
                                                  const float* __restrict__ W,
                                                  const float* __restrict__ b,
                                                  float* __restrict__ out) {
    int idx = blockIdx.x * 256 + threadIdx.x;   // 65536
    int i = idx >> 4;
    int d = idx & 15;
    float s = b[d];
#pragma unroll
    for (int k = 0; k < 8; ++k) s += diag[k * NN + i] * W[k * 16 + d];
    out[idx] = s;
}

extern "C" void kernel_launch(void* const* d_in, const int* in_sizes, int n_in,
                              void* d_out, int out_size, void* d_ws, size_t ws_size,
                              hipStream_t stream) {
    (void)in_sizes; (void)n_in; (void)out_size; (void)ws_size;
    const int*   edge_index = (const int*)d_in[0];
    const float* W_pe       = (const float*)d_in[2];
    const float* b_pe       = (const float*)d_in[3];
    float*       out        = (float*)d_out;

    const size_t MATB = (size_t)NN * NN * sizeof(float);   // 64 MB
    char* ws = (char*)d_ws;
    float* T    = (float*)(ws);
    float* P1   = (float*)(ws + MATB);
    float* P2   = (float*)(ws + 2 * MATB);
    float* diag = (float*)(ws + 3 * MATB);                  // [8][4096]

    const dim3 gemmGrid(NN / BNB, NN / BM);                 // (16, 32)

    // Build T
    zero_f4<<<(NN * NN / 4 + 255) / 256, 256, 0, stream>>>((float4*)T, NN * NN / 4);
    scatter_edges<<<NEDGE / 256, 256, 0, stream>>>(edge_index, T);
    row_normalize<<<NN, 256, 0, stream>>>(T);

    // diag(T^1)
    diag_extract<<<NN / 256, 256, 0, stream>>>(T, diag + 0 * NN);

    // T2 = T*T ; diag(T^2)
    gemm_wmma_f32<<<gemmGrid, 256, 0, stream>>>(T, T, P1);
    diag_extract<<<NN / 256, 256, 0, stream>>>(P1, diag + 1 * NN);

    // T3 = T2*T ; diag(T^3)
    gemm_wmma_f32<<<gemmGrid, 256, 0, stream>>>(P1, T, P2);
    diag_extract<<<NN / 256, 256, 0, stream>>>(P2, diag + 2 * NN);

    // diag(T^5) = <T2, T3>   (must run before P1 is overwritten)
    diag_dot<<<NN / 64, 256, 0, stream>>>(P1, P2, diag + 4 * NN);

    // T4 = T3*T ; diag(T^4)
    gemm_wmma_f32<<<gemmGrid, 256, 0, stream>>>(P2, T, P1);
    diag_extract<<<NN / 256, 256, 0, stream>>>(P1, diag + 3 * NN);

    // diag(T^6) = <T3,T3>; diag(T^7) = <T3,T4>; diag(T^8) = <T4,T4>
    diag_dot<<<NN / 64, 256, 0, stream>>>(P2, P2, diag + 5 * NN);
    diag_dot<<<NN / 64, 256, 0, stream>>>(P2, P1, diag + 6 * NN);
    diag_dot<<<NN / 64, 256, 0, stream>>>(P1, P1, diag + 7 * NN);

    // Final projection [4096,8] @ [8,16] + bias
    pe_project<<<NN * 16 / 256, 256, 0, stream>>>(diag, W_pe, b_pe, out);
}